// HeCoGATConv_2044404433794
// MI455X (gfx1250) — compile-verified
//
#include <hip/hip_runtime.h>
#include <hip/hip_bf16.h>
#include <math.h>

#define NEG_SLOPE 0.01f
#define DFEAT 64

typedef __attribute__((ext_vector_type(2))) float v2f;
typedef __attribute__((ext_vector_type(8))) float v8f;

// ---------------------------------------------------------------------------
// Order-preserving float<->uint map so segment-max can use integer atomicMax.
// ---------------------------------------------------------------------------
__device__ __forceinline__ unsigned enc_f32(float x) {
    unsigned u = __float_as_uint(x);
    return (u & 0x80000000u) ? ~u : (u | 0x80000000u);
}
__device__ __forceinline__ float dec_f32(unsigned u) {
    return (u & 0x80000000u) ? __uint_as_float(u & 0x7FFFFFFFu)
                             : __uint_as_float(~u);
}
#define ENC_NEG_INF 0x007FFFFFu   // enc_f32(-inf) = ~0xFF800000

// ---------------------------------------------------------------------------
// Init: zero out[ndst*64], zero denom[ndst], set emax_enc[ndst] = enc(-inf)
// ---------------------------------------------------------------------------
__global__ void gat_init(float* __restrict__ out, float* __restrict__ denom,
                         unsigned* __restrict__ emax_enc, int ndst) {
    int n = ndst * DFEAT;
    for (int i = blockIdx.x * blockDim.x + threadIdx.x; i < n;
         i += gridDim.x * blockDim.x) {
        out[i] = 0.0f;
        if (i < ndst) {
            denom[i] = 0.0f;
            emax_enc[i] = ENC_NEG_INF;
        }
    }
}

// ---------------------------------------------------------------------------
// el/er GEMV via V_WMMA_F32_16X16X4_F32 (wave32).
// One wave computes dot(feat[row], attn) for 16 consecutive rows.
//   A (16x4, f32, 2 VGPR): lanes 0-15 hold row M=lane, K={kb,kb+1};
//                          lanes 16-31 hold row M=lane-16, K={kb+2,kb+3}.
//   B (4x16): attn slice replicated across all 16 columns (mirrors A layout).
//   D (16x16, 8 VGPR): D[m][n] = el[row0+m] for every n.
//     lane 0  holds el[row0+0..7]  in acc[0..7]
//     lane 16 holds el[row0+8..15] in acc[0..7]
// Tail handling: clamp row index (redundant read of a valid row); the store
// remains range-guarded. This keeps every load unconditional so EXEC is never
// touched inside the accumulation loop (WMMA requires EXEC all-ones anyway).
// ---------------------------------------------------------------------------
__global__ void gat_rowdot_wmma(const float* __restrict__ feat,
                                const float* __restrict__ attn,
                                float* __restrict__ out, int nrows) {
    int gtid  = blockIdx.x * blockDim.x + threadIdx.x;
    int wave  = gtid >> 5;
    int lane  = threadIdx.x & 31;
    int row0  = wave * 16;
    if (row0 >= nrows) return;              // wave-uniform exit: EXEC stays full

    int khalf = lane >> 4;                  // 0: K={0,1}, 1: K={2,3} within step
    int m     = lane & 15;
    int row   = row0 + m;
    if (row >= nrows) row = nrows - 1;      // clamp: value never stored

    // v2f element streams, pre-offset by this lane-half's K position.
    const v2f* fr = (const v2f*)(feat + (size_t)row * DFEAT) + khalf;
    const v2f* at = (const v2f*)attn + khalf;

    v8f acc = {0.f, 0.f, 0.f, 0.f, 0.f, 0.f, 0.f, 0.f};
#pragma unroll
    for (int kb = 0; kb < DFEAT / 4; ++kb) {            // 16 K-steps of 4
        v2f a = fr[kb * 2];                             // floats kb*4 + khalf*2
        v2f b = at[kb * 2];
        acc = __builtin_amdgcn_wmma_f32_16x16x4_f32(
            /*neg_a=*/false, a, /*neg_b=*/false, b,
            /*c_mod=*/(short)0, acc, /*reuse_a=*/false, /*reuse_b=*/false);
    }

    if ((lane & 15) == 0) {                 // lanes 0 and 16
        int base = row0 + (khalf ? 8 : 0);
#pragma unroll
        for (int i = 0; i < 8; ++i) {
            int r = base + i;
            if (r < nrows) out[r] = acc[i];
        }
    }
}

// ---------------------------------------------------------------------------
// Pass 1 over edges: e = leaky_relu(el[src]+er[dst]); segment max by dst.
// ---------------------------------------------------------------------------
__global__ void gat_edge_max(const int* __restrict__ src,
                             const int* __restrict__ dst,
                             const float* __restrict__ el,
                             const float* __restrict__ er,
                             float* __restrict__ e_out,
                             unsigned* __restrict__ emax_enc, int E) {
    int i = blockIdx.x * blockDim.x + threadIdx.x;
    if (i >= E) return;
    int s = src[i], d = dst[i];
    float e = el[s] + er[d];
    e = (e >= 0.0f) ? e : NEG_SLOPE * e;
    e_out[i] = e;
    atomicMax(&emax_enc[d], enc_f32(e));
}

// ---------------------------------------------------------------------------
// Pass 2 over edges: w = exp(e - emax[dst]) (in place over e); denom[dst]+=w.
// ---------------------------------------------------------------------------
__global__ void gat_edge_expsum(const int* __restrict__ dst,
                                float* __restrict__ ew,         // e in, w out
                                const unsigned* __restrict__ emax_enc,
                                float* __restrict__ denom, int E) {
    int i = blockIdx.x * blockDim.x + threadIdx.x;
    if (i >= E) return;
    int d = dst[i];
    float x = expf(ew[i] - dec_f32(emax_enc[d]));
    ew[i] = x;
    atomicAdd(&denom[d], x);
}

// ---------------------------------------------------------------------------
// Pass 3 over edges: out[dst,:] += (w/denom[dst]) * feat_src[src,:].
// 64 threads per edge -> coalesced row gather + coalesced f32 atomics (L2).
// ---------------------------------------------------------------------------
__global__ void gat_aggregate(const int* __restrict__ src,
                              const int* __restrict__ dst,
                              const float* __restrict__ feat_src,
                              const float* __restrict__ w,
                              const float* __restrict__ denom,
                              float* __restrict__ out, int E) {
    int t = blockIdx.x * blockDim.x + threadIdx.x;
    int e = t >> 6;             // 64 lanes per edge
    int c = t & 63;
    if (e >= E) return;
    int s = src[e], d = dst[e];
    float a = w[e] / denom[d];
    atomicAdd(&out[(size_t)d * DFEAT + c],
              a * feat_src[(size_t)s * DFEAT + c]);
}

// ---------------------------------------------------------------------------
extern "C" void kernel_launch(void* const* d_in, const int* in_sizes, int n_in,
                              void* d_out, int out_size, void* d_ws, size_t ws_size,
                              hipStream_t stream) {
    const float* feat_src = (const float*)d_in[0];
    const float* feat_dst = (const float*)d_in[1];
    const float* attn_l   = (const float*)d_in[2];
    const float* attn_r   = (const float*)d_in[3];
    const int*   src      = (const int*)d_in[4];
    const int*   dst      = (const int*)d_in[5];
    float*       out      = (float*)d_out;

    const int n_src = in_sizes[0] / DFEAT;
    const int n_dst = in_sizes[1] / DFEAT;
    const int E     = in_sizes[4];

    // Workspace carve-up (floats): el | er | e/w | denom | emax_enc
    float*    el       = (float*)d_ws;
    float*    er       = el + n_src;
    float*    ew       = er + n_dst;          // e, overwritten with w
    float*    denom    = ew + E;
    unsigned* emax_enc = (unsigned*)(denom + n_dst);

    const int BLK = 256;

    // Init out/denom/emax
    {
        int n = n_dst * DFEAT;
        int grid = (n + BLK - 1) / BLK;
        gat_init<<<grid, BLK, 0, stream>>>(out, denom, emax_enc, n_dst);
    }

    // el / er via WMMA (16 rows per wave)
    {
        int waves_s = (n_src + 15) / 16;
        int grid_s  = (waves_s * 32 + BLK - 1) / BLK;
        gat_rowdot_wmma<<<grid_s, BLK, 0, stream>>>(feat_src, attn_l, el, n_src);
        int waves_d = (n_dst + 15) / 16;
        int grid_d  = (waves_d * 32 + BLK - 1) / BLK;
        gat_rowdot_wmma<<<grid_d, BLK, 0, stream>>>(feat_dst, attn_r, er, n_dst);
    }

    // Edge score + segment max
    {
        int grid = (E + BLK - 1) / BLK;
        gat_edge_max<<<grid, BLK, 0, stream>>>(src, dst, el, er, ew, emax_enc, E);
    }

    // exp + segment sum
    {
        int grid = (E + BLK - 1) / BLK;
        gat_edge_expsum<<<grid, BLK, 0, stream>>>(dst, ew, emax_enc, denom, E);
    }

    // Weighted aggregation
    {
        long long work = (long long)E * DFEAT;
        int grid = (int)((work + BLK - 1) / BLK);
        gat_aggregate<<<grid, BLK, 0, stream>>>(src, dst, feat_src, ew, denom, out, E);
    }
}